// MultiTaskGNN_27891517620239
// MI455X (gfx1250) — compile-verified
//
#include <hip/hip_runtime.h>
#include <math.h>

typedef float v2f __attribute__((ext_vector_type(2)));
typedef float v8f __attribute__((ext_vector_type(8)));

#define HD 4
#define CH 32
#define FDIM 128   // HD*CH

// ---------------- helpers ----------------
__device__ __forceinline__ void atomicMaxF(float* addr, float val) {
  // monotonic float<->int bit trick; addr must be initialized to a float (-inf here)
  if (val >= 0.0f) {
    atomicMax((int*)addr, __float_as_int(val));
  } else {
    atomicMin((unsigned int*)addr, (unsigned int)__float_as_int(val));
  }
}

__device__ __forceinline__ float leaky02(float a) {
  return a > 0.0f ? a : 0.2f * a;
}

// ---------------- generic ----------------
__global__ void k_zero(float* p, int count) {
  int i = blockIdx.x * blockDim.x + threadIdx.x;
  if (i < count) p[i] = 0.0f;
}

// x[n,9] @ W1[9,128] -> out[n,128]
__global__ void k_linear1(const float* __restrict__ x, const float* __restrict__ W1,
                          float* __restrict__ out, int n) {
  int idx = blockIdx.x * blockDim.x + threadIdx.x;
  if (idx >= n * FDIM) return;
  int nid = idx >> 7, j = idx & (FDIM - 1);
  float acc = 0.0f;
#pragma unroll
  for (int k = 0; k < 9; ++k) acc += x[nid * 9 + k] * W1[k * FDIM + j];
  out[idx] = acc;
}

// per (node, head): s = <h, a_src>, d = <h, a_dst>; also init amax/denom
__global__ void k_attn_sd(const float* __restrict__ h, const float* __restrict__ a_src,
                          const float* __restrict__ a_dst, float* __restrict__ s,
                          float* __restrict__ d, float* __restrict__ amax,
                          float* __restrict__ den, int n) {
  int idx = blockIdx.x * blockDim.x + threadIdx.x;
  if (idx >= n * HD) return;
  int nid = idx >> 2, hd = idx & 3;
  const float* hp = h + (size_t)nid * FDIM + hd * CH;
  const float* ap = a_src + hd * CH;
  const float* dp = a_dst + hd * CH;
  float ss = 0.0f, dd = 0.0f;
#pragma unroll
  for (int c = 0; c < CH; ++c) {
    float v = hp[c];
    ss += v * ap[c];
    dd += v * dp[c];
  }
  s[idx] = ss;
  d[idx] = dd;
  amax[idx] = -INFINITY;
  den[idx] = 0.0f;
}

// segment max of leaky_relu(s[src]+d[dst]) into amax[dst]   (float4 = b128 loads)
__global__ void k_edge_max(const int* __restrict__ ei, int E, int n,
                           const float4* __restrict__ s4, const float4* __restrict__ d4,
                           float* __restrict__ amax) {
  int e = blockIdx.x * blockDim.x + threadIdx.x;
  int Etot = E + n;
  if (e >= Etot) return;
  int src = (e < E) ? ei[e] : (e - E);
  int dst = (e < E) ? ei[E + e] : (e - E);
  float4 sv = s4[src];
  float4 dv = d4[dst];
  atomicMaxF(&amax[dst * HD + 0], leaky02(sv.x + dv.x));
  atomicMaxF(&amax[dst * HD + 1], leaky02(sv.y + dv.y));
  atomicMaxF(&amax[dst * HD + 2], leaky02(sv.z + dv.z));
  atomicMaxF(&amax[dst * HD + 3], leaky02(sv.w + dv.w));
}

// segment sum of exp(alpha - amax[dst]) into den[dst]
__global__ void k_edge_sum(const int* __restrict__ ei, int E, int n,
                           const float4* __restrict__ s4, const float4* __restrict__ d4,
                           const float4* __restrict__ amax4, float* __restrict__ den) {
  int e = blockIdx.x * blockDim.x + threadIdx.x;
  int Etot = E + n;
  if (e >= Etot) return;
  int src = (e < E) ? ei[e] : (e - E);
  int dst = (e < E) ? ei[E + e] : (e - E);
  float4 sv = s4[src];
  float4 dv = d4[dst];
  float4 mv = amax4[dst];
  atomicAdd(&den[dst * HD + 0], expf(leaky02(sv.x + dv.x) - mv.x));
  atomicAdd(&den[dst * HD + 1], expf(leaky02(sv.y + dv.y) - mv.y));
  atomicAdd(&den[dst * HD + 2], expf(leaky02(sv.z + dv.z) - mv.z));
  atomicAdd(&den[dst * HD + 3], expf(leaky02(sv.w + dv.w) - mv.w));
}

// wave-per-edge weighted aggregation: agg[dst] += h[src] * w  (lane = channel)
__global__ void k_edge_agg(const int* __restrict__ ei, int E, int n,
                           const float4* __restrict__ s4, const float4* __restrict__ d4,
                           const float4* __restrict__ amax4, const float4* __restrict__ den4,
                           const float* __restrict__ h, float* __restrict__ agg) {
  int gid = blockIdx.x * blockDim.x + threadIdx.x;
  int e = gid >> 5;
  int lane = gid & 31;
  int Etot = E + n;
  if (e >= Etot) return;
  int src = (e < E) ? ei[e] : (e - E);
  int dst = (e < E) ? ei[E + e] : (e - E);
  float4 sv = s4[src];
  float4 dv = d4[dst];
  float4 mv = amax4[dst];
  float4 nv = den4[dst];
  float w[HD];
  w[0] = expf(leaky02(sv.x + dv.x) - mv.x) / nv.x;
  w[1] = expf(leaky02(sv.y + dv.y) - mv.y) / nv.y;
  w[2] = expf(leaky02(sv.z + dv.z) - mv.z) / nv.z;
  w[3] = expf(leaky02(sv.w + dv.w) - mv.w) / nv.w;
  const float* hp = h + (size_t)src * FDIM;
  float* op = agg + (size_t)dst * FDIM;
#pragma unroll
  for (int hd = 0; hd < HD; ++hd) {
    atomicAdd(&op[hd * CH + lane], hp[hd * CH + lane] * w[hd]);
  }
}

// mean over heads + bias + relu
__global__ void k_headmean(const float* __restrict__ agg, const float* __restrict__ b,
                           float* __restrict__ out, int n) {
  int idx = blockIdx.x * blockDim.x + threadIdx.x;
  if (idx >= n * CH) return;
  int nid = idx >> 5, c = idx & 31;
  const float* ap = agg + (size_t)nid * FDIM;
  float v = (ap[c] + ap[CH + c] + ap[2 * CH + c] + ap[3 * CH + c]) * 0.25f + b[c];
  out[idx] = v > 0.0f ? v : 0.0f;
}

// ---------------- WMMA f32 GEMM: h1[N,32] @ W2[32,128] -> out[N,128] ----------------
// block = 256 threads = 8 waves; each wave computes a 16x16 output tile (16 nodes x 16 cols)
__global__ void k_linear2_wmma(const float* __restrict__ h1, const float* __restrict__ W2,
                               float* __restrict__ out, int n) {
  __shared__ float At[16 * 32];
  int tid = threadIdx.x;
  int wave = tid >> 5;
  int lane = tid & 31;
  int m0 = blockIdx.x * 16;

  for (int i = tid; i < 16 * 32; i += 256) {
    int r = i >> 5, c = i & 31;
    int rr = m0 + r;
    if (rr >= n) rr = n - 1;  // clamp (keeps EXEC full for WMMA)
    At[i] = h1[(size_t)rr * 32 + c];
  }
  __syncthreads();

  int col = wave * 16 + (lane & 15);
  int koff = (lane < 16) ? 0 : 2;  // A/B K split across lane halves (ISA 7.12.2)
  int row = lane & 15;

  v8f acc = {};
#pragma unroll
  for (int kk = 0; kk < 8; ++kk) {
    int k0 = kk * 4;
    v2f a, b;
    a.x = At[row * 32 + k0 + koff];
    a.y = At[row * 32 + k0 + koff + 1];
    b.x = W2[(k0 + koff) * FDIM + col];
    b.y = W2[(k0 + koff + 1) * FDIM + col];
    acc = __builtin_amdgcn_wmma_f32_16x16x4_f32(false, a, false, b, (short)0, acc,
                                                false, false);
  }

  int mbase = (lane < 16) ? 0 : 8;  // C/D: VGPR v holds rows v and v+8
  float* op = out + (size_t)(m0 + mbase) * FDIM + col;
  if (m0 + 16 <= n) {
    // uniform fast path: whole tile in range, branchless stores
#pragma unroll
    for (int v = 0; v < 8; ++v) op[(size_t)v * FDIM] = acc[v];
  } else {
#pragma unroll
    for (int v = 0; v < 8; ++v) {
      if (m0 + mbase + v < n) op[(size_t)v * FDIM] = acc[v];
    }
  }
}

// node head: h2[N,32] @ Wn[32,2] + bn
__global__ void k_node_logits(const float* __restrict__ h2, const float* __restrict__ Wn,
                              const float* __restrict__ bn, float* __restrict__ out, int n) {
  int idx = blockIdx.x * blockDim.x + threadIdx.x;
  if (idx >= n * 2) return;
  int nid = idx >> 1, j = idx & 1;
  float acc = bn[j];
#pragma unroll
  for (int c = 0; c < CH; ++c) acc += h2[(size_t)nid * CH + c] * Wn[c * 2 + j];
  out[idx] = acc;
}

// ---------------- edge MLP head with WMMA ----------------
// block = 128 threads = 4 waves; 16 label-edges per block.
// ei16x64 = [h2[a] | h2[b]]; hidden = relu(ei @ We1 + be1); logit = hidden . We2 + be2
__global__ void k_edge_mlp_wmma(const float* __restrict__ h2, const int* __restrict__ eli,
                                const float* __restrict__ We1, const float* __restrict__ be1,
                                const float* __restrict__ We2, const float* __restrict__ be2,
                                float* __restrict__ out, int EL) {
  __shared__ float Et[16 * 64];
  __shared__ float accRow[16];
  int tid = threadIdx.x;
  int wave = tid >> 5;
  int lane = tid & 31;
  int e0 = blockIdx.x * 16;

  for (int i = tid; i < 16 * 64; i += 128) {
    int r = i >> 6, c = i & 63;
    int ee = e0 + r;
    if (ee >= EL) ee = EL - 1;  // clamp, stores guarded below
    int node = (c < 32) ? eli[ee] : eli[EL + ee];
    Et[i] = h2[(size_t)node * 32 + (c & 31)];
  }
  if (tid < 16) accRow[tid] = 0.0f;
  __syncthreads();

  int col = wave * 16 + (lane & 15);
  int koff = (lane < 16) ? 0 : 2;
  int row = lane & 15;

  v8f acc = {};
#pragma unroll
  for (int kk = 0; kk < 16; ++kk) {
    int k0 = kk * 4;
    v2f a, b;
    a.x = Et[row * 64 + k0 + koff];
    a.y = Et[row * 64 + k0 + koff + 1];
    b.x = We1[(k0 + koff) * 64 + col];
    b.y = We1[(k0 + koff + 1) * 64 + col];
    acc = __builtin_amdgcn_wmma_f32_16x16x4_f32(false, a, false, b, (short)0, acc,
                                                false, false);
  }

  float w2 = We2[col];
  float bias = be1[col];
  int mbase = (lane < 16) ? 0 : 8;
#pragma unroll
  for (int v = 0; v < 8; ++v) {
    float hval = acc[v] + bias;
    hval = hval > 0.0f ? hval : 0.0f;  // relu
    atomicAdd(&accRow[mbase + v], hval * w2);  // ds_add_f32 row reduction
  }
  __syncthreads();
  if (e0 + 16 <= EL) {
    if (tid < 16) out[e0 + tid] = accRow[tid] + be2[0];
  } else {
    if (tid < 16 && (e0 + tid) < EL) out[e0 + tid] = accRow[tid] + be2[0];
  }
}

// ---------------- host ----------------
static inline int cdiv_i(long a, long b) { return (int)((a + b - 1) / b); }

extern "C" void kernel_launch(void* const* d_in, const int* in_sizes, int n_in,
                              void* d_out, int out_size, void* d_ws, size_t ws_size,
                              hipStream_t stream) {
  (void)n_in; (void)out_size; (void)ws_size;

  const float* x   = (const float*)d_in[0];
  const int*   ei  = (const int*)d_in[1];
  const int*   eli = (const int*)d_in[2];
  const float* W1  = (const float*)d_in[3];
  const float* as1 = (const float*)d_in[4];
  const float* ad1 = (const float*)d_in[5];
  const float* b1  = (const float*)d_in[6];
  const float* W2  = (const float*)d_in[7];
  const float* as2 = (const float*)d_in[8];
  const float* ad2 = (const float*)d_in[9];
  const float* b2  = (const float*)d_in[10];
  const float* Wn  = (const float*)d_in[11];
  const float* bn  = (const float*)d_in[12];
  const float* We1 = (const float*)d_in[13];
  const float* be1 = (const float*)d_in[14];
  const float* We2 = (const float*)d_in[15];
  const float* be2 = (const float*)d_in[16];

  int n  = in_sizes[0] / 9;
  int E  = in_sizes[1] / 2;
  int EL = in_sizes[2] / 2;
  int Etot = E + n;

  float* ws   = (float*)d_ws;
  float* htmp = ws; ws += (size_t)n * FDIM;   // 16B-aligned sub-buffers (sizes are multiples of 4 floats)
  float* agg  = ws; ws += (size_t)n * FDIM;
  float* s    = ws; ws += (size_t)n * HD;
  float* dv   = ws; ws += (size_t)n * HD;
  float* amax = ws; ws += (size_t)n * HD;
  float* den  = ws; ws += (size_t)n * HD;
  float* h1   = ws; ws += (size_t)n * CH;
  float* h2   = ws; ws += (size_t)n * CH;

  const float4* s4  = (const float4*)s;
  const float4* d4  = (const float4*)dv;
  const float4* m4  = (const float4*)amax;
  const float4* n4  = (const float4*)den;

  float* node_out = (float*)d_out;
  float* edge_out = node_out + (size_t)n * 2;

  dim3 b256(256);

  // ---- layer 1 ----
  k_linear1<<<cdiv_i((long)n * FDIM, 256), b256, 0, stream>>>(x, W1, htmp, n);
  k_attn_sd<<<cdiv_i((long)n * HD, 256), b256, 0, stream>>>(htmp, as1, ad1, s, dv, amax, den, n);
  k_zero<<<cdiv_i((long)n * FDIM, 256), b256, 0, stream>>>(agg, n * FDIM);
  k_edge_max<<<cdiv_i(Etot, 256), b256, 0, stream>>>(ei, E, n, s4, d4, amax);
  k_edge_sum<<<cdiv_i(Etot, 256), b256, 0, stream>>>(ei, E, n, s4, d4, m4, den);
  k_edge_agg<<<cdiv_i((long)Etot * 32, 256), b256, 0, stream>>>(ei, E, n, s4, d4, m4, n4, htmp, agg);
  k_headmean<<<cdiv_i((long)n * CH, 256), b256, 0, stream>>>(agg, b1, h1, n);

  // ---- layer 2 (WMMA GEMM) ----
  k_linear2_wmma<<<cdiv_i(n, 16), b256, 0, stream>>>(h1, W2, htmp, n);
  k_attn_sd<<<cdiv_i((long)n * HD, 256), b256, 0, stream>>>(htmp, as2, ad2, s, dv, amax, den, n);
  k_zero<<<cdiv_i((long)n * FDIM, 256), b256, 0, stream>>>(agg, n * FDIM);
  k_edge_max<<<cdiv_i(Etot, 256), b256, 0, stream>>>(ei, E, n, s4, d4, amax);
  k_edge_sum<<<cdiv_i(Etot, 256), b256, 0, stream>>>(ei, E, n, s4, d4, m4, den);
  k_edge_agg<<<cdiv_i((long)Etot * 32, 256), b256, 0, stream>>>(ei, E, n, s4, d4, m4, n4, htmp, agg);
  k_headmean<<<cdiv_i((long)n * CH, 256), b256, 0, stream>>>(agg, b2, h2, n);

  // ---- heads ----
  k_node_logits<<<cdiv_i((long)n * 2, 256), b256, 0, stream>>>(h2, Wn, bn, node_out, n);
  k_edge_mlp_wmma<<<cdiv_i(EL, 16), dim3(128), 0, stream>>>(h2, eli, We1, be1, We2, be2, edge_out, EL);
}